// ST_AffineDiff_24111946400123
// MI455X (gfx1250) — compile-verified
//
#include <hip/hip_runtime.h>
#include <math.h>

// Problem constants (match reference)
#define NB 128
#define CC 3
#define HH 256
#define WW 256
#define MAXNORM 5.371920351148152f
#define MAX_SQ 16

typedef __attribute__((ext_vector_type(2))) float v2f;
typedef __attribute__((ext_vector_type(8))) float v8f;

// ---------------------------------------------------------------------------
// WMMA helpers: 16x16 f32 tiles in LDS (row-major), products via chained
// V_WMMA_F32_16X16X4_F32 (K split into 4 chunks of 4).
//
// A-matrix 16x4 f32 layout (ISA 7.12.2): lane l holds row M=l&15;
//   VGPR v holds K = v + 2*(l>>4). Chunk q offsets K by 4q.
// B-matrix 4x16 f32: rows striped across lanes within a VGPR; mirror of A
//   with M->N: lane l holds col N=l&15, VGPR v holds K = 4q + v + 2*(l>>4).
// C/D 16x16 f32: lane l, VGPR r -> (M = r + 8*(l>>4), N = l&15).
// ---------------------------------------------------------------------------
__device__ __forceinline__ v2f ld_afrag(const float* T, int lane, int q) {
  int m  = lane & 15;
  int kb = 4 * q + 2 * (lane >> 4);
  v2f a;
  a.x = T[m * 16 + kb];
  a.y = T[m * 16 + kb + 1];
  return a;
}

__device__ __forceinline__ v2f ld_bfrag(const float* T, int lane, int q) {
  int n  = lane & 15;
  int kb = 4 * q + 2 * (lane >> 4);
  v2f b;
  b.x = T[kb * 16 + n];
  b.y = T[(kb + 1) * 16 + n];
  return b;
}

__device__ __forceinline__ v8f tile_mm(const float* X, const float* Y, int lane) {
  v8f acc = {};
#pragma unroll
  for (int q = 0; q < 4; ++q) {
    v2f a = ld_afrag(X, lane, q);
    v2f b = ld_bfrag(Y, lane, q);
    acc = __builtin_amdgcn_wmma_f32_16x16x4_f32(false, a, false, b,
                                                (short)0, acc, false, false);
  }
  return acc;
}

__device__ __forceinline__ void st_frag(float* T, int lane, v8f c) {
  int n  = lane & 15;
  int mb = 8 * (lane >> 4);
#pragma unroll
  for (int r = 0; r < 8; ++r) T[(mb + r) * 16 + n] = c[r];
}

// ---------------------------------------------------------------------------
// Kernel A: batched matrix exponential of 128 padded 3x3 affines.
// One wave (32 threads) per block; each wave owns 5 matrices packed
// block-diagonally into a 16x16 tile (rows/cols 15 zero).
// ---------------------------------------------------------------------------
__global__ void __launch_bounds__(32)
expm_kernel(const float* __restrict__ theta, float* __restrict__ thout) {
  __shared__ float TA[256], T2[256], T4[256], T6[256], TP[256], TU[256], TV[256];
  __shared__ float sA[5][9];
  __shared__ int   sk[5];

  const int lane = threadIdx.x;
  const int blk  = blockIdx.x;

  // --- per-matrix scaling (lanes 0..4) ---
  if (lane < 5) {
    const int mat = blk * 5 + lane;
    float a[9];
#pragma unroll
    for (int i = 0; i < 9; ++i) a[i] = 0.f;
    if (mat < NB) {
#pragma unroll
      for (int i = 0; i < 6; ++i) a[i] = theta[mat * 6 + i];
    }
    float ss = 0.f;
#pragma unroll
    for (int i = 0; i < 9; ++i) ss += a[i] * a[i];
    const float fro = sqrtf(ss);
    float ns = ceilf(log2f(fro / MAXNORM));   // fro==0 -> -inf -> clamped below
    ns = fmaxf(ns, 0.f);
    ns = fminf(ns, (float)MAX_SQ);
    const float sc = exp2f(-ns);
#pragma unroll
    for (int i = 0; i < 9; ++i) sA[lane][i] = a[i] * sc;
    sk[lane] = (int)ns;
  }
  __syncthreads();

  // --- assemble block-diagonal Asc tile ---
#pragma unroll
  for (int j = 0; j < 8; ++j) {
    const int idx = lane * 8 + j;
    const int r = idx >> 4, c = idx & 15;
    float v = 0.f;
    if (r < 15 && c < 15 && (r / 3) == (c / 3))
      v = sA[r / 3][(r % 3) * 3 + (c % 3)];
    TA[idx] = v;
  }
  __syncthreads();

  // --- Pade-13 powers ---
  v8f f;
  f = tile_mm(TA, TA, lane); __syncthreads(); st_frag(T2, lane, f); __syncthreads();
  f = tile_mm(T2, T2, lane); __syncthreads(); st_frag(T4, lane, f); __syncthreads();
  f = tile_mm(T4, T2, lane); __syncthreads(); st_frag(T6, lane, f); __syncthreads();

  const float b0  = 6.476475253248e16f, b1  = 3.238237626624e16f;
  const float b2  = 7771770303897600.f, b3  = 1187353796428800.f;
  const float b4  = 129060195264000.f,  b5  = 10559470521600.f;
  const float b6  = 670442572800.f,     b7  = 33522128640.f;
  const float b8  = 1323241920.f,       b9  = 40840800.f;
  const float b10 = 960960.f,           b11 = 16380.f;
  const float b12 = 182.f,              b13 = 1.f;

  // TP = b13*A6 + b11*A4 + b9*A2
#pragma unroll
  for (int j = 0; j < 8; ++j) {
    const int idx = lane * 8 + j;
    TP[idx] = b13 * T6[idx] + b11 * T4[idx] + b9 * T2[idx];
  }
  __syncthreads();
  f = tile_mm(T6, TP, lane);                 // A6 @ (...)
  __syncthreads(); st_frag(TU, lane, f); __syncthreads();
  // TU += b7*A6 + b5*A4 + b3*A2 + b1*I
#pragma unroll
  for (int j = 0; j < 8; ++j) {
    const int idx = lane * 8 + j;
    const int r = idx >> 4, c = idx & 15;
    float v = TU[idx] + b7 * T6[idx] + b5 * T4[idx] + b3 * T2[idx];
    if (r == c && r < 15) v += b1;
    TU[idx] = v;
  }
  __syncthreads();
  f = tile_mm(TA, TU, lane);                 // U = A @ (...)
  __syncthreads(); st_frag(TP, lane, f); __syncthreads();   // TP = U

  // TU (scratch) = b12*A6 + b10*A4 + b8*A2
#pragma unroll
  for (int j = 0; j < 8; ++j) {
    const int idx = lane * 8 + j;
    TU[idx] = b12 * T6[idx] + b10 * T4[idx] + b8 * T2[idx];
  }
  __syncthreads();
  f = tile_mm(T6, TU, lane);                 // A6 @ (...)
  __syncthreads(); st_frag(TV, lane, f); __syncthreads();
  // TV += b6*A6 + b4*A4 + b2*A2 + b0*I   -> TV = V
#pragma unroll
  for (int j = 0; j < 8; ++j) {
    const int idx = lane * 8 + j;
    const int r = idx >> 4, c = idx & 15;
    float v = TV[idx] + b6 * T6[idx] + b4 * T4[idx] + b2 * T2[idx];
    if (r == c && r < 15) v += b0;
    TV[idx] = v;
  }
  __syncthreads();

  // --- per-matrix solve (U+V) R = (V-U), then squaring; lanes 0..4 ---
  if (lane < 5) {
    const int mat = blk * 5 + lane;
    if (mat < NB) {
      const int rb = lane * 3;
      float M[9], Nn[9];
#pragma unroll
      for (int i = 0; i < 3; ++i)
#pragma unroll
        for (int j2 = 0; j2 < 3; ++j2) {
          const float u = TP[(rb + i) * 16 + rb + j2];
          const float v = TV[(rb + i) * 16 + rb + j2];
          M[i * 3 + j2]  = u + v;
          Nn[i * 3 + j2] = v - u;
        }
      // 3x3 inverse via adjugate
      const float c00 =  M[4] * M[8] - M[5] * M[7];
      const float c01 = -(M[3] * M[8] - M[5] * M[6]);
      const float c02 =  M[3] * M[7] - M[4] * M[6];
      const float det = M[0] * c00 + M[1] * c01 + M[2] * c02;
      const float id  = 1.f / det;
      float inv[9];
      inv[0] =  c00 * id;
      inv[1] = -(M[1] * M[8] - M[2] * M[7]) * id;
      inv[2] =  (M[1] * M[5] - M[2] * M[4]) * id;
      inv[3] =  c01 * id;
      inv[4] =  (M[0] * M[8] - M[2] * M[6]) * id;
      inv[5] = -(M[0] * M[5] - M[2] * M[3]) * id;
      inv[6] =  c02 * id;
      inv[7] = -(M[0] * M[7] - M[1] * M[6]) * id;
      inv[8] =  (M[0] * M[4] - M[1] * M[3]) * id;

      float R[9];
#pragma unroll
      for (int i = 0; i < 3; ++i)
#pragma unroll
        for (int j2 = 0; j2 < 3; ++j2)
          R[i * 3 + j2] = inv[i * 3 + 0] * Nn[0 * 3 + j2]
                        + inv[i * 3 + 1] * Nn[1 * 3 + j2]
                        + inv[i * 3 + 2] * Nn[2 * 3 + j2];

      const int k = sk[lane];
      for (int it = 0; it < k; ++it) {     // data-dependent squaring (k <= 16)
        float S[9];
#pragma unroll
        for (int i = 0; i < 3; ++i)
#pragma unroll
          for (int j2 = 0; j2 < 3; ++j2)
            S[i * 3 + j2] = R[i * 3 + 0] * R[0 * 3 + j2]
                          + R[i * 3 + 1] * R[1 * 3 + j2]
                          + R[i * 3 + 2] * R[2 * 3 + j2];
#pragma unroll
        for (int i = 0; i < 9; ++i) R[i] = S[i];
      }
#pragma unroll
      for (int i = 0; i < 6; ++i) thout[mat * 6 + i] = R[i];  // rows 0..1
    }
  }
}

// ---------------------------------------------------------------------------
// Kernel B: fused affine_grid + bilinear grid_sample (zeros padding,
// align_corners=True). One thread per (n,h,w); 3 channels per thread.
// Block = one output row (256 threads = 8 waves); theta uniform per block.
// Bandwidth-bound: ~200 MiB traffic -> ~9us floor at 23.3 TB/s.
// ---------------------------------------------------------------------------
__global__ void __launch_bounds__(256)
sample_kernel(const float* __restrict__ x, const float* __restrict__ th,
              float* __restrict__ out) {
  const int w = threadIdx.x;
  const int h = blockIdx.y;
  const int n = blockIdx.z;

  const float t00 = th[n * 6 + 0], t01 = th[n * 6 + 1], t02 = th[n * 6 + 2];
  const float t10 = th[n * 6 + 3], t11 = th[n * 6 + 4], t12 = th[n * 6 + 5];

  const float gx = (2.0f * (float)w) / (float)(WW - 1) - 1.0f;
  const float gy = (2.0f * (float)h) / (float)(HH - 1) - 1.0f;

  const float sx = t00 * gx + t01 * gy + t02;
  const float sy = t10 * gx + t11 * gy + t12;

  const float ix = (sx + 1.0f) * (0.5f * (float)(WW - 1));
  const float iy = (sy + 1.0f) * (0.5f * (float)(HH - 1));

  const float fx0 = floorf(ix), fy0 = floorf(iy);
  const int ix0 = (int)fx0, iy0 = (int)fy0;
  const int ix1 = ix0 + 1,  iy1 = iy0 + 1;

  const float wx1 = ix - fx0, wx0 = 1.0f - wx1;
  const float wy1 = iy - fy0, wy0 = 1.0f - wy1;

  const bool vx0 = (ix0 >= 0) && (ix0 < WW);
  const bool vx1 = (ix1 >= 0) && (ix1 < WW);
  const bool vy0 = (iy0 >= 0) && (iy0 < HH);
  const bool vy1 = (iy1 >= 0) && (iy1 < HH);

  const int cx0 = ix0 < 0 ? 0 : (ix0 > WW - 1 ? WW - 1 : ix0);
  const int cx1 = ix1 < 0 ? 0 : (ix1 > WW - 1 ? WW - 1 : ix1);
  const int cy0 = iy0 < 0 ? 0 : (iy0 > HH - 1 ? HH - 1 : iy0);
  const int cy1 = iy1 < 0 ? 0 : (iy1 > HH - 1 ? HH - 1 : iy1);

  const float w00 = (vy0 && vx0) ? wy0 * wx0 : 0.0f;
  const float w01 = (vy0 && vx1) ? wy0 * wx1 : 0.0f;
  const float w10 = (vy1 && vx0) ? wy1 * wx0 : 0.0f;
  const float w11 = (vy1 && vx1) ? wy1 * wx1 : 0.0f;

  const int o00 = cy0 * WW + cx0, o01 = cy0 * WW + cx1;
  const int o10 = cy1 * WW + cx0, o11 = cy1 * WW + cx1;

  const float* __restrict__ xp = x + (size_t)n * CC * HH * WW;
  float* __restrict__ op = out + (size_t)n * CC * HH * WW + (size_t)h * WW + w;

#pragma unroll
  for (int c = 0; c < CC; ++c) {
    const float* __restrict__ p = xp + c * (HH * WW);
    const float v = w00 * p[o00] + w01 * p[o01] + w10 * p[o10] + w11 * p[o11];
    op[(size_t)c * (HH * WW)] = v;
  }
}

// ---------------------------------------------------------------------------
extern "C" void kernel_launch(void* const* d_in, const int* in_sizes, int n_in,
                              void* d_out, int out_size, void* d_ws, size_t ws_size,
                              hipStream_t stream) {
  const float* x     = (const float*)d_in[0];   // [N,C,H,W] f32
  const float* theta = (const float*)d_in[1];   // [N,6] f32
  float* out   = (float*)d_out;                 // [N,C,H,W] f32
  float* thout = (float*)d_ws;                  // [N,6] expm result

  // 26 waves x 5 matrices = 130 slots >= 128
  expm_kernel<<<dim3(26), dim3(32), 0, stream>>>(theta, thout);
  sample_kernel<<<dim3(1, HH, NB), dim3(WW), 0, stream>>>(x, thout, out);
}